// mLSTMCell_33990371180672
// MI455X (gfx1250) — compile-verified
//
#include <hip/hip_runtime.h>
#include <hip/hip_bf16.h>

// mLSTM single-step cell for MI455X (gfx1250, wave32).
// B=64, S=1, F=4096, H=16, HD=256.
// Outputs concatenated flat: c_new (B*H*HD*HD), n_new (B*H*HD), m_new (B*H), h_norm (B*F).

typedef float v2f __attribute__((ext_vector_type(2)));
typedef float v8f __attribute__((ext_vector_type(8)));

#define NB    64
#define NH    16
#define NF    4096
#define NHD   256
#define NK3   12288

// ---------------------------------------------------------------------------
// 0) zero the gate accumulators in workspace (harness poisons ws with 0xAA)
// ---------------------------------------------------------------------------
__global__ void zero_ws_kernel(float* __restrict__ p, int n) {
    int i = blockIdx.x * blockDim.x + threadIdx.x;
    if (i < n) p[i] = 0.0f;
}

// ---------------------------------------------------------------------------
// 1) gate GEMMs via V_WMMA_F32_16X16X4_F32:
//    i_tilde/f_tilde = qkv(64x12288) @ W(12288x16), K split into 24 chunks
//    of 512 (8 chunks per q/k/v source so a chunk never crosses a concat
//    boundary). One wave per (gate, mtile, chunk) = 192 waves; partial 16x16
//    D tiles combined into ws with global_atomic_add_f32.
//
//    A-frag layout (16x4 f32, ISA 7.12.2): lane<16 holds A[m=lane][k0+0/1] in
//    v0/v1, lane>=16 holds A[m=lane-16][k0+2/3]. B-frag (4x16) mirrored:
//    lane<16 holds B[k0+0/1][n=lane], lane>=16 holds B[k0+2/3][n=lane-16].
//    D layout: vgpr r, lane l -> D[r + 8*(l>=16)][l%16].
// ---------------------------------------------------------------------------
__global__ __launch_bounds__(32) void gate_wmma_kernel(
    const float* __restrict__ q, const float* __restrict__ k,
    const float* __restrict__ v,
    const float* __restrict__ wi, const float* __restrict__ wf,
    float* __restrict__ gacc /* [2][64][16] */)
{
    const int bx    = blockIdx.x;      // 0..191
    const int gate  = bx / 96;         // 0 = input gate, 1 = forget gate
    const int rem   = bx % 96;
    const int mtile = rem / 24;        // 0..3  (rows of 16 batches)
    const int chunk = rem % 24;        // 0..23 (K chunks of 512)
    const int tens  = chunk / 8;       // 0=q, 1=k, 2=v
    const int koff  = (chunk % 8) * 512;

    const float* __restrict__ src = (tens == 0) ? q : (tens == 1) ? k : v;
    const float* __restrict__ w   = (gate == 0) ? wi : wf;

    const int lane = threadIdx.x;      // 0..31
    const int ml   = lane & 15;
    const int ksel = (lane >> 4) * 2;  // 0 or 2

    const int brow = mtile * 16 + ml;  // batch index this lane feeds for A
    const float* __restrict__ arow  = src + (size_t)brow * NF + koff;
    const float* __restrict__ wbase = w + ((size_t)(tens * NF + koff)) * NH + ml;

    v8f acc = {0.f, 0.f, 0.f, 0.f, 0.f, 0.f, 0.f, 0.f};
    for (int kc = 0; kc < 512; kc += 4) {
        v2f a, b;
        a.x = arow[kc + ksel];
        a.y = arow[kc + ksel + 1];
        b.x = wbase[(size_t)(kc + ksel) * NH];
        b.y = wbase[(size_t)(kc + ksel + 1) * NH];
        acc = __builtin_amdgcn_wmma_f32_16x16x4_f32(
            /*neg_a=*/false, a, /*neg_b=*/false, b,
            /*c_mod=*/(short)0, acc, /*reuse_a=*/false, /*reuse_b=*/false);
    }

    // scatter-add partial D into gacc[gate][batch][h]
    const int ncol  = ml;
    const int mbase = mtile * 16 + (lane >> 4) * 8;
    float* __restrict__ g = gacc + gate * (NB * NH);
#pragma unroll
    for (int r = 0; r < 8; ++r) {
        atomicAdd(&g[(mbase + r) * NH + ncol], acc[r]);
    }
}

// ---------------------------------------------------------------------------
// 2) gate finalize: softplus/max/exp per (b,h); writes m_new to d_out and
//    i_prime/f_prime to workspace.
// ---------------------------------------------------------------------------
__global__ void gate_finalize_kernel(
    const float* __restrict__ gacc,
    const float* __restrict__ wi_bias, const float* __restrict__ wf_bias,
    const float* __restrict__ m_in,
    float* __restrict__ ip_out, float* __restrict__ fp_out,
    float* __restrict__ m_out)
{
    const int idx = blockIdx.x * blockDim.x + threadIdx.x; // b*16 + h
    if (idx >= NB * NH) return;
    const int h = idx & (NH - 1);

    const float it = gacc[idx] + wi_bias[h];
    const float ft = gacc[NB * NH + idx] + wf_bias[h];
    // log_f = -softplus(-ft), numerically stable
    const float sp    = fmaxf(-ft, 0.0f) + log1pf(expf(-fabsf(ft)));
    const float log_f = -sp;
    const float m_old = m_in[idx];
    const float m_new = fmaxf(log_f + m_old, it);

    ip_out[idx] = expf(it - m_new);
    fp_out[idx] = expf(log_f + m_old - m_new);
    m_out[idx]  = m_new;
}

// ---------------------------------------------------------------------------
// 3) fused bandwidth kernel: one block per (b,h) 256x256 tile of c.
//    thread t owns columns 4*(t%64)..+3, row-phase r = t/64.
//    Streams c once (float4 coalesced), writes c_new once, and accumulates
//    nominator[j] = sum_i q[i]*c_new[i][j] on the fly. Then n_new, the
//    q.n_new reduction, and the per-head layernorm — all in-block.
// ---------------------------------------------------------------------------
__device__ __forceinline__ float block_sum256(float x, float* red) {
    const int t = threadIdx.x;
    red[t] = x;
    __syncthreads();
#pragma unroll
    for (int s = 128; s > 0; s >>= 1) {
        if (t < s) red[t] += red[t + s];
        __syncthreads();
    }
    const float r = red[0];
    __syncthreads();
    return r;
}

__global__ __launch_bounds__(256) void mlstm_update_kernel(
    const float* __restrict__ q, const float* __restrict__ k,
    const float* __restrict__ v, const float* __restrict__ c,
    const float* __restrict__ n_in, const float* __restrict__ ln_scale,
    const float* __restrict__ ip_arr, const float* __restrict__ fp_arr,
    const float* __restrict__ mn_arr,
    float* __restrict__ c_out, float* __restrict__ n_out,
    float* __restrict__ h_out)
{
    __shared__ float q_s[NHD];
    __shared__ float ks_s[NHD];
    __shared__ float v_s[NHD];
    __shared__ float nom_s[4][NHD];
    __shared__ float red[256];

    const int bh = blockIdx.x;        // 0..1023
    const int b  = bh >> 4;
    const int h  = bh & 15;
    const int t  = threadIdx.x;

    const float* __restrict__ qh  = q + (size_t)b * NF + h * NHD;
    const float* __restrict__ kh  = k + (size_t)b * NF + h * NHD;
    const float* __restrict__ vh  = v + (size_t)b * NF + h * NHD;
    const float* __restrict__ cin = c + (size_t)bh * (NHD * NHD);
    float* __restrict__ cout      = c_out + (size_t)bh * (NHD * NHD);

    const float ip = ip_arr[bh];
    const float fp = fp_arr[bh];
    const float mn = mn_arr[bh];

    q_s[t]  = qh[t];
    ks_s[t] = kh[t] * 0.0625f;   // 1/sqrt(256)
    v_s[t]  = vh[t];
    __syncthreads();

    const int cc = t & 63;       // column group: columns 4cc..4cc+3
    const int r  = t >> 6;       // row phase 0..3

    const float4 vj = *(const float4*)&v_s[4 * cc];
    float4 nom = make_float4(0.f, 0.f, 0.f, 0.f);

#pragma unroll 4
    for (int ii = 0; ii < 64; ++ii) {
        const int i   = 4 * ii + r;
        const float a = ip * ks_s[i];
        const float qi = q_s[i];
        const float4 cold = *(const float4*)(cin + (size_t)i * NHD + 4 * cc);
        if (ii + 2 < 64)  // emits global_prefetch_b8 (gfx1250)
            __builtin_prefetch(cin + (size_t)(i + 8) * NHD + 4 * cc, 0, 0);
        float4 cn;
        cn.x = fp * cold.x + a * vj.x;
        cn.y = fp * cold.y + a * vj.y;
        cn.z = fp * cold.z + a * vj.z;
        cn.w = fp * cold.w + a * vj.w;
        *(float4*)(cout + (size_t)i * NHD + 4 * cc) = cn;
        nom.x += qi * cn.x;
        nom.y += qi * cn.y;
        nom.z += qi * cn.z;
        nom.w += qi * cn.w;
    }

    nom_s[r][4 * cc + 0] = nom.x;
    nom_s[r][4 * cc + 1] = nom.y;
    nom_s[r][4 * cc + 2] = nom.z;
    nom_s[r][4 * cc + 3] = nom.w;
    __syncthreads();

    // column j = t
    const float nomj = nom_s[0][t] + nom_s[1][t] + nom_s[2][t] + nom_s[3][t];

    // n_new[j] = f'*n[j] + i'*k_scaled[j]
    const float nn = fp * n_in[(size_t)bh * NHD + t] + ip * ks_s[t];
    n_out[(size_t)bh * NHD + t] = nn;

    // denominator = max(|q . n_new|, exp(-m_new))
    const float qdn = block_sum256(q_s[t] * nn, red);
    const float den = fmaxf(fabsf(qdn), expf(-mn)) + 1e-6f;
    const float ht  = nomj / den;

    // per-head layernorm over HD (scale-only, eps=1e-6)
    const float mu  = block_sum256(ht, red) * (1.0f / NHD);
    const float d   = ht - mu;
    const float var = block_sum256(d * d, red) * (1.0f / NHD);
    const float hn  = d * rsqrtf(var + 1e-6f) * ln_scale[h * NHD + t];

    h_out[(size_t)b * NF + h * NHD + t] = hn;
}

// ---------------------------------------------------------------------------
// launch
// ---------------------------------------------------------------------------
extern "C" void kernel_launch(void* const* d_in, const int* in_sizes, int n_in,
                              void* d_out, int out_size, void* d_ws, size_t ws_size,
                              hipStream_t stream) {
    const float* q       = (const float*)d_in[0];
    const float* k       = (const float*)d_in[1];
    const float* v       = (const float*)d_in[2];
    const float* c       = (const float*)d_in[3];
    const float* n       = (const float*)d_in[4];
    const float* m       = (const float*)d_in[5];
    const float* wi_ker  = (const float*)d_in[6];
    const float* wi_bias = (const float*)d_in[7];
    const float* wf_ker  = (const float*)d_in[8];
    const float* wf_bias = (const float*)d_in[9];
    const float* ln      = (const float*)d_in[10];

    float* out   = (float*)d_out;
    float* c_out = out;                                        // 64*16*256*256
    float* n_out = c_out + (size_t)NB * NH * NHD * NHD;        // 64*16*256
    float* m_out = n_out + (size_t)NB * NH * NHD;              // 64*16
    float* h_out = m_out + (size_t)NB * NH;                    // 64*4096

    float* ws   = (float*)d_ws;
    float* gacc = ws;                 // 2 * 64 * 16 = 2048 floats
    float* ip   = ws + 2048;          // 1024
    float* fp   = ws + 3072;          // 1024

    zero_ws_kernel<<<8, 256, 0, stream>>>(gacc, 2 * NB * NH);
    gate_wmma_kernel<<<192, 32, 0, stream>>>(q, k, v, wi_ker, wf_ker, gacc);
    gate_finalize_kernel<<<4, 256, 0, stream>>>(gacc, wi_bias, wf_bias, m, ip, fp, m_out);
    mlstm_update_kernel<<<NB * NH, 256, 0, stream>>>(q, k, v, c, n, ln, ip, fp, m_out,
                                                     c_out, n_out, h_out);
}